// ToChoices_66494683676757
// MI455X (gfx1250) — compile-verified
//
#include <hip/hip_runtime.h>

// Problem constants (match the reference).
#define B_      64
#define H_      256
#define W_      256
#define C_      3
#define HWC_    (H_ * W_ * C_)        // 196608 floats per sample per source
#define OUTPB_  (H_ * W_ * 2 * C_)    // 393216 floats per sample out
#define PIX_PER_THREAD 4              // 4 pixels -> 48B in per source, 96B out (16B aligned)

typedef __attribute__((ext_vector_type(4))) float v4f;

// out[b,h,w,j,c] = (idx[b,j]==0 ? fakes : reals)[b,h,w,c]
// Each thread: load 3x b128 from srcA, 3x b128 from srcB, emit 6x b128 NT stores.
__global__ __launch_bounds__(256) void to_choices_kernel(
    const float* __restrict__ reals,
    const float* __restrict__ fakes,
    const long long* __restrict__ shuffle_indices,
    float* __restrict__ out)
{
    const int b = blockIdx.y;                              // batch sample (uniform)
    const int t = blockIdx.x * blockDim.x + threadIdx.x;   // [0, 16384): 4-pixel chunk id

    // Uniform per block -> scalarizes. Value 0 selects fakes (stack order (fakes, reals)).
    const long long i0 = shuffle_indices[2 * b + 0];
    const long long i1 = shuffle_indices[2 * b + 1];
    const float* __restrict__ srcA = (i0 == 0 ? fakes : reals) + (size_t)b * HWC_;
    const float* __restrict__ srcB = (i1 == 0 ? fakes : reals) + (size_t)b * HWC_;

    // 4 pixels = 12 input floats per source (3 x float4), 24 output floats (6 x float4).
    const v4f* __restrict__ pa = (const v4f*)(srcA + (size_t)t * 12);
    const v4f* __restrict__ pb = (const v4f*)(srcB + (size_t)t * 12);
    v4f* __restrict__ po = (v4f*)(out + (size_t)b * OUTPB_ + (size_t)t * 24);

    // RT loads: inputs (100.7 MB total) stay resident in the 192 MB L2 across replays.
    const v4f A0 = pa[0], A1 = pa[1], A2 = pa[2];
    const v4f B0 = pb[0], B1 = pb[1], B2 = pb[2];

    // A = a0..a11 (pixels p0:a0-2, p1:a3-5, p2:a6-8, p3:a9-11), B likewise.
    // Output per pixel: [a3k, a3k+1, a3k+2, b3k, b3k+1, b3k+2]
    const v4f o0 = {A0.x, A0.y, A0.z, B0.x};   // a0 a1 a2 b0
    const v4f o1 = {B0.y, B0.z, A0.w, A1.x};   // b1 b2 a3 a4
    const v4f o2 = {A1.y, B0.w, B1.x, B1.y};   // a5 b3 b4 b5
    const v4f o3 = {A1.z, A1.w, A2.x, B1.z};   // a6 a7 a8 b6
    const v4f o4 = {B1.w, B2.x, A2.y, A2.z};   // b7 b8 a9 a10
    const v4f o5 = {A2.w, B2.y, B2.z, B2.w};   // a11 b9 b10 b11

    // NT stores: 100.7 MB output stream must not evict the L2-resident inputs.
    __builtin_nontemporal_store(o0, po + 0);
    __builtin_nontemporal_store(o1, po + 1);
    __builtin_nontemporal_store(o2, po + 2);
    __builtin_nontemporal_store(o3, po + 3);
    __builtin_nontemporal_store(o4, po + 4);
    __builtin_nontemporal_store(o5, po + 5);
}

extern "C" void kernel_launch(void* const* d_in, const int* in_sizes, int n_in,
                              void* d_out, int out_size, void* d_ws, size_t ws_size,
                              hipStream_t stream) {
    const float*     reals = (const float*)d_in[0];
    const float*     fakes = (const float*)d_in[1];
    const long long* sidx  = (const long long*)d_in[2];   // int64 (B, 2)
    float*           out   = (float*)d_out;

    // 65536 pixels/sample / 4 per thread = 16384 threads/sample = 64 blocks of 256.
    dim3 grid(64, B_, 1);
    dim3 block(256, 1, 1);
    to_choices_kernel<<<grid, block, 0, stream>>>(reals, fakes, sidx, out);
}